// SpeciesGeneGNN_46256797778370
// MI455X (gfx1250) — compile-verified
//
#include <hip/hip_runtime.h>

#define HD 128
#define HDP 132              // padded LDS row stride (bank-conflict-free)
#define ED 64
#define WAVE 32
#define MT 2                 // M tiles per block
#define ROWS (MT * 16)       // 32 rows per block

typedef float v2f __attribute__((ext_vector_type(2)));
typedef float v8f __attribute__((ext_vector_type(8)));

// ---------------------------------------------------------------------------
// fp32 WMMA 16x16x4 step.  Layouts per CDNA5 ISA 7.12.2:
//   A (16x4): lane l -> row (l&15); comp c -> K = 2*(l>>4)+c
//   B (4x16): lane l -> col (l&15); comp c -> K = 2*(l>>4)+c
//   C/D     : vgpr r, lane l -> row r + 8*(l>>4), col (l&15)
// ---------------------------------------------------------------------------
__device__ __forceinline__ v8f wmma_step(v2f a, v2f b, v8f acc) {
  return __builtin_amdgcn_wmma_f32_16x16x4_f32(
      /*neg_a=*/false, a, /*neg_b=*/false, b,
      /*c_mod=*/(short)0, acc, /*reuse_a=*/false, /*reuse_b=*/false);
}

// out[N, HD] = X[N, K] @ W[K, HD] + b   (fp32 WMMA, LDS-staged A, 2 M-tiles)
template <int K>
__global__ void linear_bias_kernel(const float* __restrict__ X,
                                   const float* __restrict__ W,
                                   const float* __restrict__ b,
                                   float* __restrict__ out, int Nrows) {
  constexpr int KP = K + 4;
  __shared__ float sX[ROWS * KP];
  const int m_base = blockIdx.x * ROWS;

  for (int i = threadIdx.x; i < ROWS * (K / 4); i += blockDim.x) {
    const int row = i / (K / 4);
    const int col = (i % (K / 4)) * 4;
    int gr = m_base + row; if (gr >= Nrows) gr = Nrows - 1;   // clamp, no divergence
    *(float4*)(sX + row * KP + col) = *(const float4*)(X + (size_t)gr * K + col);
  }
  __syncthreads();

  const int wave = threadIdx.x / WAVE;
  const int lane = threadIdx.x % WAVE;
  const int h = lane >> 4;
  const int lm = lane & 15;
  const int n = wave * 16 + lm;     // 8 waves cover all HD=128 cols

  v8f acc0 = {}, acc1 = {};
#pragma unroll
  for (int k0 = 0; k0 < K; k0 += 4) {
    const int ka = k0 + 2 * h;
    v2f w; w.x = W[(size_t)ka * HD + n]; w.y = W[(size_t)(ka + 1) * HD + n];
    const v2f a0 = *(const v2f*)(sX + lm * KP + ka);
    const v2f a1 = *(const v2f*)(sX + (16 + lm) * KP + ka);
    acc0 = wmma_step(a0, w, acc0);
    acc1 = wmma_step(a1, w, acc1);
  }
  const float bias = b[n];
#pragma unroll
  for (int r = 0; r < 8; ++r) {
    const int row0 = m_base + r + 8 * h;
    if (row0 < Nrows) out[(size_t)row0 * HD + n] = acc0[r] + bias;
    const int row1 = m_base + 16 + r + 8 * h;
    if (row1 < Nrows) out[(size_t)row1 * HD + n] = acc1[r] + bias;
  }
}

// Edge scatter: agg[dst] += xsrc[src]; cnt[dst] += 1.  32 lanes/edge, float4.
__global__ void scatter_accum_kernel(const float* __restrict__ xsrc,
                                     const int* __restrict__ src,
                                     const int* __restrict__ dst,
                                     float* __restrict__ agg,
                                     float* __restrict__ cnt, int E) {
  const long long t = (long long)blockIdx.x * blockDim.x + threadIdx.x;
  const int e = (int)(t >> 5);
  const int c = (int)(t & 31);
  if (e >= E) return;
  const int s = src[e];
  const int d = dst[e];
  const float4 v = *(const float4*)(xsrc + (size_t)s * HD + c * 4);
  float* ag = agg + (size_t)d * HD + c * 4;
  atomicAdd(ag + 0, v.x);
  atomicAdd(ag + 1, v.y);
  atomicAdd(ag + 2, v.z);
  atomicAdd(ag + 3, v.w);
  if (c == 0) atomicAdd(cnt + d, 1.0f);
}

__global__ void zero_kernel(float* __restrict__ p, size_t n) {
  const size_t i = (size_t)blockIdx.x * blockDim.x + threadIdx.x;
  if (i < n) p[i] = 0.0f;
}

// out = relu(scale * ( (agg1/max(cnt1,1))@Wl1 + bl1 + xdst@Wr1
//                    [+ (agg2/max(cnt2,1))@Wl2 + bl2 + xdst@Wr2 ] ))
// A-sources staged (pre-scaled) in LDS once per block; 2 M-tiles per block.
template <bool HAS2>
__global__ void sage_combine_kernel(
    const float* __restrict__ agg1, const float* __restrict__ cnt1,
    const float* __restrict__ Wl1, const float* __restrict__ bl1,
    const float* __restrict__ Wr1,
    const float* __restrict__ agg2, const float* __restrict__ cnt2,
    const float* __restrict__ Wl2, const float* __restrict__ bl2,
    const float* __restrict__ Wr2,
    const float* __restrict__ xdst,
    float* __restrict__ out, int Nrows, float scale) {
  __shared__ float sA1[ROWS * HDP];
  __shared__ float sXd[ROWS * HDP];
  __shared__ float sA2[HAS2 ? ROWS * HDP : 4];

  const int m_base = blockIdx.x * ROWS;

  for (int i = threadIdx.x; i < ROWS * (HD / 4); i += blockDim.x) {
    const int row = i >> 5;          // HD/4 == 32 float4 per row
    const int col = (i & 31) * 4;
    int gr = m_base + row; if (gr >= Nrows) gr = Nrows - 1;
    const float ic1 = 1.0f / fmaxf(cnt1[gr], 1.0f);
    float4 v = *(const float4*)(agg1 + (size_t)gr * HD + col);
    v.x *= ic1; v.y *= ic1; v.z *= ic1; v.w *= ic1;
    *(float4*)(sA1 + row * HDP + col) = v;
    *(float4*)(sXd + row * HDP + col) = *(const float4*)(xdst + (size_t)gr * HD + col);
    if constexpr (HAS2) {
      const float ic2 = 1.0f / fmaxf(cnt2[gr], 1.0f);
      float4 u = *(const float4*)(agg2 + (size_t)gr * HD + col);
      u.x *= ic2; u.y *= ic2; u.z *= ic2; u.w *= ic2;
      *(float4*)(sA2 + row * HDP + col) = u;
    }
  }
  __syncthreads();

  const int wave = threadIdx.x / WAVE;
  const int lane = threadIdx.x % WAVE;
  const int h = lane >> 4;
  const int lm = lane & 15;
  const int n = wave * 16 + lm;

  v8f acc0 = {}, acc1 = {};
#pragma unroll
  for (int k0 = 0; k0 < HD; k0 += 4) {
    const int ka = k0 + 2 * h;
    {
      v2f w; w.x = Wl1[(size_t)ka * HD + n]; w.y = Wl1[(size_t)(ka + 1) * HD + n];
      const v2f a0 = *(const v2f*)(sA1 + lm * HDP + ka);
      const v2f a1 = *(const v2f*)(sA1 + (16 + lm) * HDP + ka);
      acc0 = wmma_step(a0, w, acc0);
      acc1 = wmma_step(a1, w, acc1);
    }
    {
      v2f w; w.x = Wr1[(size_t)ka * HD + n]; w.y = Wr1[(size_t)(ka + 1) * HD + n];
      const v2f x0 = *(const v2f*)(sXd + lm * HDP + ka);
      const v2f x1 = *(const v2f*)(sXd + (16 + lm) * HDP + ka);
      acc0 = wmma_step(x0, w, acc0);
      acc1 = wmma_step(x1, w, acc1);
    }
    if constexpr (HAS2) {
      {
        v2f w; w.x = Wl2[(size_t)ka * HD + n]; w.y = Wl2[(size_t)(ka + 1) * HD + n];
        const v2f a0 = *(const v2f*)(sA2 + lm * HDP + ka);
        const v2f a1 = *(const v2f*)(sA2 + (16 + lm) * HDP + ka);
        acc0 = wmma_step(a0, w, acc0);
        acc1 = wmma_step(a1, w, acc1);
      }
      {
        v2f w; w.x = Wr2[(size_t)ka * HD + n]; w.y = Wr2[(size_t)(ka + 1) * HD + n];
        const v2f x0 = *(const v2f*)(sXd + lm * HDP + ka);
        const v2f x1 = *(const v2f*)(sXd + (16 + lm) * HDP + ka);
        acc0 = wmma_step(x0, w, acc0);
        acc1 = wmma_step(x1, w, acc1);
      }
    }
  }

  float bias = bl1[n];
  if constexpr (HAS2) bias += bl2[n];
#pragma unroll
  for (int r = 0; r < 8; ++r) {
    const int row0 = m_base + r + 8 * h;
    if (row0 < Nrows)
      out[(size_t)row0 * HD + n] = fmaxf((acc0[r] + bias) * scale, 0.0f);
    const int row1 = m_base + 16 + r + 8 * h;
    if (row1 < Nrows)
      out[(size_t)row1 * HD + n] = fmaxf((acc1[r] + bias) * scale, 0.0f);
  }
}

extern "C" void kernel_launch(void* const* d_in, const int* in_sizes, int n_in,
                              void* d_out, int out_size, void* d_ws, size_t ws_size,
                              hipStream_t stream) {
  const float* species_emb = (const float*)d_in[0];
  const float* gene_emb    = (const float*)d_in[1];
  const float* Ws          = (const float*)d_in[2];
  const float* bs          = (const float*)d_in[3];
  const float* Wg          = (const float*)d_in[4];
  const float* bg          = (const float*)d_in[5];
  const float* Wl          = (const float*)d_in[6];
  const float* bl          = (const float*)d_in[7];
  const float* Wr          = (const float*)d_in[8];
  const int* hg_src  = (const int*)d_in[9];
  const int* hg_dst  = (const int*)d_in[10];
  const int* bt_src  = (const int*)d_in[11];
  const int* bt_dst  = (const int*)d_in[12];
  const int* sim_src = (const int*)d_in[13];
  const int* sim_dst = (const int*)d_in[14];

  const int NS    = in_sizes[0] / ED;
  const int NG    = in_sizes[1] / ED;
  const int L     = in_sizes[6] / (3 * HD * HD);
  const int E_HG  = in_sizes[9];
  const int E_SIM = in_sizes[13];

  const size_t gsz = (size_t)NG * HD;
  const size_t ssz = (size_t)NS * HD;

  float* ws = (float*)d_ws;
  float* gxA  = ws; ws += gsz;
  float* gxB  = ws; ws += gsz;
  float* sxA  = ws; ws += ssz;
  float* sxB  = ws; ws += ssz;
  float* aggA = ws; ws += gsz;   // gene <- species (has_gene)
  float* aggC = ws; ws += gsz;   // gene <- gene    (similar_to)
  float* aggB = ws; ws += ssz;   // species <- gene (belongs_to)
  float* cntA = ws; ws += NG;
  float* cntC = ws; ws += NG;
  float* cntB = ws; ws += NS;
  const size_t zero_len = 2 * gsz + ssz + 2 * (size_t)NG + (size_t)NS;
  float* zero_base = aggA;

  const dim3 blk(256);

  // Input linears
  linear_bias_kernel<ED><<<dim3((NS + ROWS - 1) / ROWS), blk, 0, stream>>>(
      species_emb, Ws, bs, sxA, NS);
  linear_bias_kernel<ED><<<dim3((NG + ROWS - 1) / ROWS), blk, 0, stream>>>(
      gene_emb, Wg, bg, gxA, NG);

  float* gx_cur = gxA; float* gx_nxt = gxB;
  float* sx_cur = sxA; float* sx_nxt = sxB;
  float* sx_final = (float*)d_out;
  float* gx_final = (float*)d_out + ssz;

  for (int l = 0; l < L; ++l) {
    zero_kernel<<<dim3((unsigned)((zero_len + 255) / 256)), blk, 0, stream>>>(
        zero_base, zero_len);

    const long long thg = (long long)E_HG * 32;
    scatter_accum_kernel<<<dim3((unsigned)((thg + 255) / 256)), blk, 0, stream>>>(
        sx_cur, hg_src, hg_dst, aggA, cntA, E_HG);
    scatter_accum_kernel<<<dim3((unsigned)((thg + 255) / 256)), blk, 0, stream>>>(
        gx_cur, bt_src, bt_dst, aggB, cntB, E_HG);
    const long long tsim = (long long)E_SIM * 32;
    scatter_accum_kernel<<<dim3((unsigned)((tsim + 255) / 256)), blk, 0, stream>>>(
        gx_cur, sim_src, sim_dst, aggC, cntC, E_SIM);

    const float* Wl_l = Wl + (size_t)l * 3 * HD * HD;
    const float* bl_l = bl + (size_t)l * 3 * HD;
    const float* Wr_l = Wr + (size_t)l * 3 * HD * HD;

    float* gx_dst = (l == L - 1) ? gx_final : gx_nxt;
    float* sx_dst = (l == L - 1) ? sx_final : sx_nxt;

    // gene update: relu(0.5 * (g_from_s + g_from_g))
    sage_combine_kernel<true><<<dim3((NG + ROWS - 1) / ROWS), blk, 0, stream>>>(
        aggA, cntA, Wl_l + 0 * HD * HD, bl_l + 0 * HD, Wr_l + 0 * HD * HD,
        aggC, cntC, Wl_l + 2 * HD * HD, bl_l + 2 * HD, Wr_l + 2 * HD * HD,
        gx_cur, gx_dst, NG, 0.5f);

    // species update: relu(s_from_g)
    sage_combine_kernel<false><<<dim3((NS + ROWS - 1) / ROWS), blk, 0, stream>>>(
        aggB, cntB, Wl_l + 1 * HD * HD, bl_l + 1 * HD, Wr_l + 1 * HD * HD,
        nullptr, nullptr, nullptr, nullptr, nullptr,
        sx_cur, sx_dst, NS, 1.0f);

    if (l < L - 1) {
      float* t = gx_cur; gx_cur = gx_nxt; gx_nxt = t;
      t = sx_cur; sx_cur = sx_nxt; sx_nxt = t;
    }
  }
}